// AttentionSpatial_22574348107777
// MI455X (gfx1250) — compile-verified
//
#include <hip/hip_runtime.h>
#include <math.h>

// ---------------------------------------------------------------------------
// AttentionSpatial for MI455X (gfx1250, wave32, WMMA)
//   x,y: [1,64,64,64] f32   w_q:[64,64]  w_kv:[128,64]  w_out:[64,64]
//   temperature: [8,1,1]
// Fused flash-attention; logits are cosine-sims * temp, so |s| <= |temp[h]|:
// softmax uses the static shift m=|temp[h]| (exact, no overflow) -> no row-max.
// Row sums come from an appended ones-column in V computed by the PV WMMAs.
// ---------------------------------------------------------------------------

typedef __attribute__((ext_vector_type(2))) float v2f;
typedef __attribute__((ext_vector_type(8))) float v8f;

#define NHEADS 8
#define HD     8
#define NPIX   4096
#define CDIM   64

// ---------------------------------------------------------------------------
// Phase 1: q = Wq x, [k|v] = Wkv y; per-head L2 normalize q,k; fold temperature
// into q. q,k layout: [head][pixel][hd].  v stored TRANSPOSED: [head][hd][pixel]
// so the PV B-operand is a contiguous float2 per lane.
// ---------------------------------------------------------------------------
__global__ __launch_bounds__(128) void qkv_proj(
    const float* __restrict__ x, const float* __restrict__ y,
    const float* __restrict__ wq, const float* __restrict__ wkv,
    const float* __restrict__ temp,
    float* __restrict__ qn, float* __restrict__ kn, float* __restrict__ vt)
{
  __shared__ float sWq[CDIM * CDIM];        // 16 KB
  __shared__ float sWkv[2 * CDIM * CDIM];   // 32 KB
  const int tid = threadIdx.x;
  for (int i = tid; i < CDIM * CDIM; i += 128)     sWq[i]  = wq[i];
  for (int i = tid; i < 2 * CDIM * CDIM; i += 128) sWkv[i] = wkv[i];
  __syncthreads();

  const int p = blockIdx.x * 128 + tid;
  float acc[CDIM];

  // ---- Q = Wq @ x, normalize, * temperature ----
#pragma unroll
  for (int o = 0; o < CDIM; ++o) acc[o] = 0.f;
#pragma unroll 1
  for (int c = 0; c < CDIM; ++c) {
    const float xv = x[c * NPIX + p];
#pragma unroll
    for (int o = 0; o < CDIM; ++o) acc[o] = fmaf(sWq[o * CDIM + c], xv, acc[o]);
  }
#pragma unroll
  for (int h = 0; h < NHEADS; ++h) {
    float ss = 0.f;
#pragma unroll
    for (int d = 0; d < HD; ++d) ss = fmaf(acc[h * HD + d], acc[h * HD + d], ss);
    const float sc = temp[h] / fmaxf(sqrtf(ss), 1e-12f);
#pragma unroll
    for (int d = 0; d < HD; ++d)
      qn[((size_t)h * NPIX + p) * HD + d] = acc[h * HD + d] * sc;
  }

  // ---- K = Wkv[0:64] @ y, normalize ----
#pragma unroll
  for (int o = 0; o < CDIM; ++o) acc[o] = 0.f;
#pragma unroll 1
  for (int c = 0; c < CDIM; ++c) {
    const float yv = y[c * NPIX + p];
#pragma unroll
    for (int o = 0; o < CDIM; ++o) acc[o] = fmaf(sWkv[o * CDIM + c], yv, acc[o]);
  }
#pragma unroll
  for (int h = 0; h < NHEADS; ++h) {
    float ss = 0.f;
#pragma unroll
    for (int d = 0; d < HD; ++d) ss = fmaf(acc[h * HD + d], acc[h * HD + d], ss);
    const float sc = 1.0f / fmaxf(sqrtf(ss), 1e-12f);
#pragma unroll
    for (int d = 0; d < HD; ++d)
      kn[((size_t)h * NPIX + p) * HD + d] = acc[h * HD + d] * sc;
  }

  // ---- V = Wkv[64:128] @ y, stored transposed [head][d][pixel] ----
#pragma unroll
  for (int o = 0; o < CDIM; ++o) acc[o] = 0.f;
#pragma unroll 1
  for (int c = 0; c < CDIM; ++c) {
    const float yv = y[c * NPIX + p];
#pragma unroll
    for (int o = 0; o < CDIM; ++o)
      acc[o] = fmaf(sWkv[(CDIM + o) * CDIM + c], yv, acc[o]);
  }
#pragma unroll
  for (int h = 0; h < NHEADS; ++h)
#pragma unroll
    for (int d = 0; d < HD; ++d)
      vt[((size_t)h * HD + d) * NPIX + p] = acc[h * HD + d];
}

// ---------------------------------------------------------------------------
// Phase 2: flash attention per (head, 64-row q tile). 4 waves/block, each wave
// owns a 16-row Q tile; KV streamed in 16-row tiles.
//   S(16x16) = Q(16x8) @ K^T(8x16)       -> 2x v_wmma_f32_16x16x4_f32
//   P = exp(S - |temp|)  (exact static-shift softmax numerator)
//   [O | l](16x9) += P @ [V | 1](16x9)   -> 4x v_wmma_f32_16x16x4_f32
// P transposed (C-layout -> A-layout) via padded per-wave LDS tile.
// ---------------------------------------------------------------------------
#define PSTRIDE 20   // 16x16 tile padded to 20 floats/row: conflict-free, 8B-aligned

__global__ __launch_bounds__(128) void fa_kernel(
    const float* __restrict__ qn, const float* __restrict__ kn,
    const float* __restrict__ vt, const float* __restrict__ temp,
    float* __restrict__ ao)
{
  const int lane = threadIdx.x & 31;
  const int wid  = threadIdx.x >> 5;
  const int l16  = lane & 15;
  const int hi   = lane >> 4;              // lane half: 0 or 1
  const int head = blockIdx.x >> 6;        // 8 heads
  const int qt   = blockIdx.x & 63;        // 64 q tiles of 64 rows
  const int q0   = qt * 64 + wid * 16;     // this wave's 16 q rows

  const float* Qh = qn + (size_t)head * NPIX * HD;
  const float* Kh = kn + (size_t)head * NPIX * HD;
  const float* Vh = vt + (size_t)head * HD * NPIX;   // transposed [d][n]
  const float  mh = fabsf(temp[head]);               // exact logit bound

  // A-operand (Q): lane holds Q[l16][2*hi + {0,1}] (+4 for the second chunk)
  const v2f a0 = *(const v2f*)(Qh + (size_t)(q0 + l16) * HD + 2 * hi);
  const v2f a1 = *(const v2f*)(Qh + (size_t)(q0 + l16) * HD + 4 + 2 * hi);

  v8f O = (v8f)0.0f;                       // cols 0..7 = O, col 8 = row sum l

  __shared__ float sP[4][16 * PSTRIDE];    // 5 KB, per-wave private P tiles
  float* Pw = &sP[wid][0];

  for (int kv0 = 0; kv0 < NPIX; kv0 += 16) {
    // prefetch next K tile (global_prefetch_b8)
    __builtin_prefetch(Kh + (size_t)(kv0 + 16 + l16) * HD, 0, 3);

    // B-operand (K^T): lane holds K[kv0+l16][2*hi + {0,1}] — same layout as A
    const v2f b0 = *(const v2f*)(Kh + (size_t)(kv0 + l16) * HD + 2 * hi);
    const v2f b1 = *(const v2f*)(Kh + (size_t)(kv0 + l16) * HD + 4 + 2 * hi);

    v8f S = (v8f)0.0f;
    S = __builtin_amdgcn_wmma_f32_16x16x4_f32(false, a0, false, b0, (short)0, S, false, false);
    S = __builtin_amdgcn_wmma_f32_16x16x4_f32(false, a1, false, b1, (short)0, S, false, false);

    // P = exp(S - m); comp r of S holds row (r + 8*hi), column l16
#pragma unroll
    for (int r = 0; r < 8; ++r)
      Pw[(r + 8 * hi) * PSTRIDE + l16] = __expf(S[r] - mh);
    __syncthreads();

    // [O|l] += P @ [V|1], K=16 as 4 chunks of 4
#pragma unroll
    for (int t = 0; t < 4; ++t) {
      const v2f pa = *(const v2f*)(Pw + l16 * PSTRIDE + 4 * t + 2 * hi);
      v2f vb;
      if (l16 < HD) {
        vb = *(const v2f*)(Vh + (size_t)l16 * NPIX + kv0 + 4 * t + 2 * hi);
      } else if (l16 == HD) {
        vb.x = 1.f; vb.y = 1.f;            // ones column -> row sums
      } else {
        vb.x = 0.f; vb.y = 0.f;
      }
      O = __builtin_amdgcn_wmma_f32_16x16x4_f32(false, pa, false, vb, (short)0, O, false, false);
    }
    __syncthreads();
  }

  // epilogue: col 8 of O is the row sum; broadcast it and normalize
#pragma unroll
  for (int r = 0; r < 8; ++r) {
    const float l = __shfl(O[r], hi * 16 + HD, 32);
    if (l16 < HD) {
      const int row = q0 + r + 8 * hi;
      ao[((size_t)head * NPIX + row) * HD + l16] = O[r] / l;
    }
  }
}

// ---------------------------------------------------------------------------
// Phase 3: out = Wout @ attn_out (head-major ws -> channel-major output)
// ---------------------------------------------------------------------------
__global__ __launch_bounds__(128) void out_proj(
    const float* __restrict__ ao, const float* __restrict__ wout,
    float* __restrict__ out)
{
  __shared__ float sW[CDIM * CDIM];
  const int tid = threadIdx.x;
  for (int i = tid; i < CDIM * CDIM; i += 128) sW[i] = wout[i];
  __syncthreads();

  const int p = blockIdx.x * 128 + tid;
  float a[CDIM];
#pragma unroll
  for (int c = 0; c < CDIM; ++c) {
    const int h = c >> 3, d = c & 7;
    a[c] = ao[((size_t)h * NPIX + p) * HD + d];
  }
#pragma unroll 1
  for (int o = 0; o < CDIM; ++o) {
    float acc = 0.f;
#pragma unroll
    for (int c = 0; c < CDIM; ++c) acc = fmaf(sW[o * CDIM + c], a[c], acc);
    out[o * NPIX + p] = acc;
  }
}

// ---------------------------------------------------------------------------
extern "C" void kernel_launch(void* const* d_in, const int* in_sizes, int n_in,
                              void* d_out, int out_size, void* d_ws, size_t ws_size,
                              hipStream_t stream) {
  const float* x    = (const float*)d_in[0];
  const float* y    = (const float*)d_in[1];
  const float* wq   = (const float*)d_in[2];
  const float* wkv  = (const float*)d_in[3];
  const float* wout = (const float*)d_in[4];
  const float* temp = (const float*)d_in[5];
  float* out = (float*)d_out;

  // workspace: 4 x 1 MB
  float* qn = (float*)d_ws;                 // [8][4096][8] normalized q * temp
  float* kn = qn + NHEADS * NPIX * HD;      // [8][4096][8] normalized k
  float* vt = kn + NHEADS * NPIX * HD;      // [8][8][4096] v (transposed)
  float* ao = vt + NHEADS * NPIX * HD;      // [8][4096][8] attention output

  qkv_proj<<<dim3(NPIX / 128), dim3(128), 0, stream>>>(x, y, wq, wkv, temp, qn, kn, vt);
  fa_kernel<<<dim3(NHEADS * 64), dim3(128), 0, stream>>>(qn, kn, vt, temp, ao);
  out_proj<<<dim3(NPIX / 128), dim3(128), 0, stream>>>(ao, wout, out);
}